// SKC_65841848648481
// MI455X (gfx1250) — compile-verified
//
#include <hip/hip_runtime.h>

// MI455X (gfx1250) windowed-GEMM:
//   out[b, o*8+n, p] = sum_k W[o,k] * x[b, n*32+k, p]
// B=32, 8 windows, O=64, K=64, HW=3136 pixels, fp32 throughout.
// Memory-bound target: ~411 MB total traffic -> ~18 us @ 23.3 TB/s.
// Math on the f32 matrix pipe: V_WMMA_F32_16X16X4_F32 (exact fp32 accumulate).

typedef __attribute__((ext_vector_type(2))) float v2f;
typedef __attribute__((ext_vector_type(8))) float v8f;

#define IN_C   512
#define HW     3136          // 56*56
#define PTILES 98            // 3136 / 32 pixels per wave-tile

__global__ __launch_bounds__(256) void SKC_win_gemm_wmma_f32(
    const float* __restrict__ x,   // (32, 512, 56, 56)
    const float* __restrict__ w,   // (64, 64)
    float* __restrict__ out)       // (32, 512, 56, 56)
{
    const int lane = threadIdx.x & 31;
    const int n    = threadIdx.x >> 5;   // wave id == window id, 0..7
    const int half = lane >> 4;          // 0 | 1
    const int lm   = lane & 15;          // 0..15

    const int b     = blockIdx.x / PTILES;
    const int pt    = blockIdx.x % PTILES;
    const int pbase = pt * 32;

    const float* xb  = x + (size_t)b * IN_C * HW;
    const int    ch0 = n * 32;           // first input channel of this window

    // 8 accumulators: 4 m-tiles (M=64) x 2 n-tiles (N=32) of 16x16 f32
    v8f acc[4][2];
#pragma unroll
    for (int mt = 0; mt < 4; ++mt)
#pragma unroll
        for (int nt = 0; nt < 2; ++nt)
            acc[mt][nt] = 0.0f;

#pragma unroll 4
    for (int ks = 0; ks < 16; ++ks) {
        const int k0 = ks * 4;
        const int kk = k0 + 2 * half;    // K held by VGPR0 of A/B in this half-wave

        // A fragments: 16x4 f32 tile of W.  lane m = lm, VGPRs = K pair (kk, kk+1)
        v2f a[4];
#pragma unroll
        for (int mt = 0; mt < 4; ++mt) {
            const float* wr = w + (size_t)(mt * 16 + lm) * 64 + kk; // 8B-aligned pair
            a[mt] = *(const v2f*)wr;
        }

        // B fragments: 4x16 f32 tile of x.  VGPR v holds K = kk + v, N = lm
        v2f bf[2];
#pragma unroll
        for (int nt = 0; nt < 2; ++nt) {
            const float* xc = xb + (size_t)(ch0 + kk) * HW + (pbase + nt * 16 + lm);
            bf[nt].x = xc[0];    // channel ch0+kk
            bf[nt].y = xc[HW];   // channel ch0+kk+1
        }

        // 8 WMMAs per K-step: D = A x B + C on the f32 matrix pipe
#pragma unroll
        for (int mt = 0; mt < 4; ++mt)
#pragma unroll
            for (int nt = 0; nt < 2; ++nt)
                acc[mt][nt] = __builtin_amdgcn_wmma_f32_16x16x4_f32(
                    /*neg_a=*/false, a[mt],
                    /*neg_b=*/false, bf[nt],
                    /*c_mod=*/(short)0, acc[mt][nt],
                    /*reuse_a=*/false, /*reuse_b=*/false);
    }

    // Store: acc VGPR v, half h -> output channel o = mt*16 + v + 8*half,
    // pixel = pbase + nt*16 + lm.  out channel index = o*8 + n.
#pragma unroll
    for (int mt = 0; mt < 4; ++mt) {
#pragma unroll
        for (int nt = 0; nt < 2; ++nt) {
            const int p = pbase + nt * 16 + lm;
#pragma unroll
            for (int v = 0; v < 8; ++v) {
                const int o  = mt * 16 + v + 8 * half;
                const int oc = o * 8 + n;
                out[((size_t)b * IN_C + oc) * HW + p] = acc[mt][nt][v];
            }
        }
    }
}

extern "C" void kernel_launch(void* const* d_in, const int* in_sizes, int n_in,
                              void* d_out, int out_size, void* d_ws, size_t ws_size,
                              hipStream_t stream) {
    const float* x = (const float*)d_in[0];   // (32, 512, 56, 56) fp32
    const float* w = (const float*)d_in[1];   // (64, 64) fp32
    float* out = (float*)d_out;               // (32, 512, 56, 56) fp32

    dim3 grid(32 * PTILES);   // (batch, pixel-tile); 8 windows = 8 waves per block
    dim3 block(256);
    hipLaunchKernelGGL(SKC_win_gemm_wmma_f32, grid, block, 0, stream, x, w, out);
}